// GAT_80831284511064
// MI455X (gfx1250) — compile-verified
//
#include <hip/hip_runtime.h>
#include <hip/hip_bf16.h>
#include <math.h>

// ---- problem constants (match reference) ----
#define IN_F     256
#define HEADS    4
#define DDIM     32
#define OUT_TOT  (HEADS * DDIM)   // 128
#define NEG_SLOPE 0.2f

typedef __attribute__((ext_vector_type(2))) float v2f;
typedef __attribute__((ext_vector_type(8))) float v8f;

// -------------------------------------------------------------------------
// float atomic max via monotonic int mapping (lowers to native int atomics).
// Requires init to -inf (0xFF800000).
__device__ __forceinline__ void atomicMaxFloat(float* addr, float val) {
    if (val >= 0.0f) {
        atomicMax((int*)addr, __float_as_int(val));
    } else {
        atomicMin((unsigned int*)addr, __float_as_uint(val));
    }
}

// -------------------------------------------------------------------------
// Kernel 0: init accumulators. m -> -inf, denom -> 0, out -> 0.
__global__ void gat_init(float* __restrict__ m, float* __restrict__ denom,
                         float* __restrict__ out, int nH, int nOut) {
    int t = blockIdx.x * blockDim.x + threadIdx.x;
    if (t < nH) {
        ((unsigned int*)m)[t] = 0xFF800000u;  // -inf
        denom[t] = 0.0f;
    }
    if (t < nOut) out[t] = 0.0f;
}

// -------------------------------------------------------------------------
// Kernel 1: node transform H = X @ W^T using V_WMMA_F32_16X16X4_F32.
// Grid: ceil(N/16) blocks of 256 threads (8 waves). wave w handles
// output-feature tile w (8 tiles of 16 cover OUT_TOT=128); block b handles
// node rows [16b, 16b+16).  K=256 -> 64 wmma steps.
//
// A (16x4 f32): lanes 0-15 row M=lane, VGPR0=K0 VGPR1=K1; lanes 16-31 same
// rows, K2/K3.  B (4x16): VGPR0 holds K0(lanes0-15)/K2(lanes16-31) for
// columns N=lane&15, VGPR1 holds K1/K3.  C/D: VGPR r = row r (+8 for upper
// half-wave), col = lane&15.
//
// Tail handling: clamp A-row loads to stay in bounds (no zeroing needed —
// garbage in A-row M only affects D-row M, and the store loop guards those).
__global__ __launch_bounds__(256)
void node_gemm_wmma(const float* __restrict__ X, const float* __restrict__ W,
                    float* __restrict__ H, int N) {
    const int wave  = threadIdx.x >> 5;      // 0..7  -> feature tile
    const int lane  = threadIdx.x & 31;
    const int half  = lane >> 4;             // 0 or 1
    const int l15   = lane & 15;
    const int row   = blockIdx.x * 16 + l15;          // node index (A row)
    const int col   = wave * 16 + l15;                // output feature (B col)
    const int khalf = half * 2;                       // K offset 0 or 2

    const int rowc = row < N ? row : N - 1;           // in-bounds clamp only

    const float* __restrict__ xrow = X + (size_t)rowc * IN_F;
    const float* __restrict__ wcol = W + (size_t)col * IN_F;   // B[k][col]=W[col][k]

    __builtin_prefetch(xrow, 0, 1);
    __builtin_prefetch(wcol, 0, 1);

    v8f c = {};
    for (int k0 = 0; k0 < IN_F; k0 += 4) {
        v2f a, b;
        a.x = xrow[k0 + khalf + 0];
        a.y = xrow[k0 + khalf + 1];
        b.x = wcol[k0 + khalf + 0];
        b.y = wcol[k0 + khalf + 1];
        c = __builtin_amdgcn_wmma_f32_16x16x4_f32(
                /*neg_a=*/false, a, /*neg_b=*/false, b,
                /*c_mod=*/(short)0, c, /*reuse_a=*/false, /*reuse_b=*/false);
    }

    const int rbase = half * 8;
#pragma unroll
    for (int r = 0; r < 8; ++r) {
        int nrow = blockIdx.x * 16 + rbase + r;
        if (nrow < N) H[(size_t)nrow * OUT_TOT + col] = c[r];
    }
}

// -------------------------------------------------------------------------
// Kernel 2: per-node attention scores  s_src[n,h] = <H[n,h,:], a_src[h,:]>
__global__ void node_scores(const float* __restrict__ H,
                            const float* __restrict__ a_src,
                            const float* __restrict__ a_dst,
                            float* __restrict__ s_src,
                            float* __restrict__ s_dst, int nH) {
    int t = blockIdx.x * blockDim.x + threadIdx.x;
    if (t >= nH) return;
    int n = t >> 2, h = t & 3;
    const float* __restrict__ hv = H + (size_t)n * OUT_TOT + h * DDIM;
    const float* __restrict__ as = a_src + h * DDIM;
    const float* __restrict__ ad = a_dst + h * DDIM;
    float ss = 0.0f, sd = 0.0f;
#pragma unroll
    for (int d = 0; d < DDIM; ++d) {
        float v = hv[d];
        ss = fmaf(v, as[d], ss);
        sd = fmaf(v, ad[d], sd);
    }
    s_src[t] = ss;
    s_dst[t] = sd;
}

// -------------------------------------------------------------------------
// Kernel 3: edge logits + leaky-relu + segment max (atomic) per dst node.
__global__ void edge_logits(const float* __restrict__ ef,
                            const float* __restrict__ We,   // [HEADS]
                            const int* __restrict__ src,
                            const int* __restrict__ dst,
                            const float* __restrict__ s_src,
                            const float* __restrict__ s_dst,
                            float* __restrict__ attn,
                            float* __restrict__ m, int E) {
    int e = blockIdx.x * blockDim.x + threadIdx.x;
    if (e >= E) return;
    int s = src[e], d = dst[e];
    float x = ef[e];
    float4 ss = ((const float4*)s_src)[s];
    float4 sd = ((const float4*)s_dst)[d];
    float4 w  = ((const float4*)We)[0];
    float a0 = ss.x + sd.x + x * w.x;
    float a1 = ss.y + sd.y + x * w.y;
    float a2 = ss.z + sd.z + x * w.z;
    float a3 = ss.w + sd.w + x * w.w;
    a0 = a0 >= 0.0f ? a0 : NEG_SLOPE * a0;
    a1 = a1 >= 0.0f ? a1 : NEG_SLOPE * a1;
    a2 = a2 >= 0.0f ? a2 : NEG_SLOPE * a2;
    a3 = a3 >= 0.0f ? a3 : NEG_SLOPE * a3;
    float4 av = {a0, a1, a2, a3};
    ((float4*)attn)[e] = av;
    atomicMaxFloat(&m[d * 4 + 0], a0);
    atomicMaxFloat(&m[d * 4 + 1], a1);
    atomicMaxFloat(&m[d * 4 + 2], a2);
    atomicMaxFloat(&m[d * 4 + 3], a3);
}

// -------------------------------------------------------------------------
// Kernel 4: ex = exp(attn - m[dst]) (in place), denom[dst] += ex (atomic).
__global__ void edge_exp(const int* __restrict__ dst,
                         const float* __restrict__ m,
                         float* __restrict__ attn,      // in: logits, out: ex
                         float* __restrict__ denom, int E) {
    int e = blockIdx.x * blockDim.x + threadIdx.x;
    if (e >= E) return;
    int d = dst[e];
    float4 mv = ((const float4*)m)[d];
    float4 a  = ((const float4*)attn)[e];
    a.x = __expf(a.x - mv.x);
    a.y = __expf(a.y - mv.y);
    a.z = __expf(a.z - mv.z);
    a.w = __expf(a.w - mv.w);
    ((float4*)attn)[e] = a;
    atomicAdd(&denom[d * 4 + 0], a.x);
    atomicAdd(&denom[d * 4 + 1], a.y);
    atomicAdd(&denom[d * 4 + 2], a.z);
    atomicAdd(&denom[d * 4 + 3], a.w);
}

// -------------------------------------------------------------------------
// Kernel 5: aggregation with fused head-mean.
// One wave32 per edge; lane = output dim d.
// out[dst,d] += (1/H) * sum_h (ex/denom)[e,h] * H[src, h*32+d]
__global__ __launch_bounds__(256)
void edge_aggregate(const int* __restrict__ src, const int* __restrict__ dst,
                    const float* __restrict__ H, const float* __restrict__ ex,
                    const float* __restrict__ denom, float* __restrict__ out,
                    int E) {
    int e = blockIdx.x * 8 + (threadIdx.x >> 5);
    if (e >= E) return;
    int lane = threadIdx.x & 31;
    int s = src[e], d = dst[e];
    float4 exv = ((const float4*)ex)[e];
    float4 dn  = ((const float4*)denom)[d];
    float al0 = exv.x / dn.x * 0.25f;
    float al1 = exv.y / dn.y * 0.25f;
    float al2 = exv.z / dn.z * 0.25f;
    float al3 = exv.w / dn.w * 0.25f;
    const float* __restrict__ hs = H + (size_t)s * OUT_TOT;
    float acc = al0 * hs[lane]
              + al1 * hs[32 + lane]
              + al2 * hs[64 + lane]
              + al3 * hs[96 + lane];
    atomicAdd(&out[(size_t)d * DDIM + lane], acc);
}

// -------------------------------------------------------------------------
extern "C" void kernel_launch(void* const* d_in, const int* in_sizes, int n_in,
                              void* d_out, int out_size, void* d_ws, size_t ws_size,
                              hipStream_t stream) {
    const float* X     = (const float*)d_in[0];  // [N, 256]
    const float* ef    = (const float*)d_in[1];  // [E, 1]
    const float* Wn    = (const float*)d_in[2];  // [128, 256]
    const float* We    = (const float*)d_in[3];  // [4, 1]
    const float* a_src = (const float*)d_in[4];  // [4, 32]
    const float* a_dst = (const float*)d_in[5];  // [4, 32]
    const int*   src   = (const int*)d_in[6];    // [E]
    const int*   dst   = (const int*)d_in[7];    // [E]
    float* out = (float*)d_out;                  // [N, 32]

    const int N = in_sizes[0] / IN_F;
    const int E = in_sizes[6];

    // workspace layout
    char* ws = (char*)d_ws;
    float* H     = (float*)ws; ws += (size_t)N * OUT_TOT * sizeof(float);
    float* s_src = (float*)ws; ws += (size_t)N * HEADS * sizeof(float);
    float* s_dst = (float*)ws; ws += (size_t)N * HEADS * sizeof(float);
    float* m     = (float*)ws; ws += (size_t)N * HEADS * sizeof(float);
    float* denom = (float*)ws; ws += (size_t)N * HEADS * sizeof(float);
    float* attn  = (float*)ws; ws += (size_t)E * HEADS * sizeof(float);

    const int nH   = N * HEADS;
    const int nOut = N * DDIM;
    const int initN = nH > nOut ? nH : nOut;

    gat_init<<<(initN + 255) / 256, 256, 0, stream>>>(m, denom, out, nH, nOut);

    node_gemm_wmma<<<(N + 15) / 16, 256, 0, stream>>>(X, Wn, H, N);

    node_scores<<<(nH + 255) / 256, 256, 0, stream>>>(H, a_src, a_dst,
                                                      s_src, s_dst, nH);

    edge_logits<<<(E + 255) / 256, 256, 0, stream>>>(ef, We, src, dst,
                                                     s_src, s_dst, attn, m, E);

    edge_exp<<<(E + 255) / 256, 256, 0, stream>>>(dst, m, attn, denom, E);

    edge_aggregate<<<(E + 7) / 8, 256, 0, stream>>>(src, dst, H, attn,
                                                    denom, out, E);
}